// NeuralConnections_33474975105068
// MI455X (gfx1250) — compile-verified
//
#include <hip/hip_runtime.h>
#include <hip/hip_bf16.h>

#define N_NEURONS 4096
#define HIDDEN 16

typedef __attribute__((ext_vector_type(2))) float v2f;
typedef __attribute__((ext_vector_type(4))) float f32x4;
typedef __attribute__((ext_vector_type(8))) float v8f;

// ---------------------------------------------------------------------------
// Prologue: A[n,h] = feat[n,:] @ W1[0:3,h]          (16x16 tile per wave)
//           Bb[n,h] = feat[n,:] @ W1[3:6,h] + b1[h]
// One wave32 per 16 rows; two V_WMMA_F32_16X16X4_F32 (shared A operand).
// K dimension padded 3 -> 4 with zeros.
// ---------------------------------------------------------------------------
__global__ __launch_bounds__(32) void nc_prologue_wmma(
    const float* __restrict__ feat, const float* __restrict__ W1,
    const float* __restrict__ b1, float* __restrict__ Aout,
    float* __restrict__ Bout) {
  const int lane = threadIdx.x;        // 0..31, EXEC all ones
  const int m    = lane & 15;          // M (rows) / N (cols) index
  const bool hi  = lane >= 16;         // K-half selector
  const int base = blockIdx.x * 16;    // neuron-row base of this tile

  // A-matrix 16x4 (M x K): lanes 0-15 -> v0=K0,v1=K1 ; lanes 16-31 -> v0=K2,v1=K3(pad 0)
  const float* frow = feat + (base + m) * 3;
  v2f amat;
  amat.x = hi ? frow[2] : frow[0];
  amat.y = hi ? 0.0f    : frow[1];

  // B-matrix 4x16 (K x N): same half-split striping; row K=3 is zero pad.
  v2f bA, bB;
  bA.x = hi ? W1[2 * HIDDEN + m] : W1[0 * HIDDEN + m];
  bA.y = hi ? 0.0f               : W1[1 * HIDDEN + m];
  bB.x = hi ? W1[5 * HIDDEN + m] : W1[3 * HIDDEN + m];
  bB.y = hi ? 0.0f               : W1[4 * HIDDEN + m];

  // C layout: VGPR r, lane l -> (M = r + 8*(l>=16), N = l&15). b1 indexes N,
  // so broadcasting b1[n] into all 8 C registers adds the bias per column.
  v8f c0 = {};
  const float bn = b1[m];
  v8f cb = {bn, bn, bn, bn, bn, bn, bn, bn};

  v8f dA = __builtin_amdgcn_wmma_f32_16x16x4_f32(
      false, amat, false, bA, (short)0, c0, false, false);
  v8f dB = __builtin_amdgcn_wmma_f32_16x16x4_f32(
      false, amat, false, bB, (short)0, cb, false, false);

  const int rowoff = base + (hi ? 8 : 0);
#pragma unroll
  for (int r = 0; r < 8; ++r) {
    Aout[(rowoff + r) * HIDDEN + m] = dA[r];
    Bout[(rowoff + r) * HIDDEN + m] = dB[r];
  }
}

// ---------------------------------------------------------------------------
// Main N^2 kernel: out[i,j] = b2 + sum_k W2[k] * max(A[i,k] + Bb[j,k], 0)
// Block = 256 threads -> 32i x 32j tile; thread = 1 i x 4 j (float4 NT store).
// Bandwidth-bound on the 67 MB output write; A/Bb (512 KB) stay L2-resident.
// ---------------------------------------------------------------------------
__global__ __launch_bounds__(256) void nc_pair_kernel(
    const float* __restrict__ A, const float* __restrict__ Bb,
    const float* __restrict__ W2, const float* __restrict__ b2,
    float* __restrict__ out) {
  const int tx = threadIdx.x & 7;   // j-quad within tile
  const int ty = threadIdx.x >> 3;  // i within tile
  const int i  = blockIdx.y * 32 + ty;
  const int j0 = blockIdx.x * 32 + tx * 4;

  // Per-thread A row (128-bit loads; broadcast across the 8 threads sharing i)
  const f32x4* arow = (const f32x4*)(A + i * HIDDEN);
  f32x4 a0 = arow[0], a1 = arow[1], a2 = arow[2], a3 = arow[3];
  float a[16] = {a0.x, a0.y, a0.z, a0.w, a1.x, a1.y, a1.z, a1.w,
                 a2.x, a2.y, a2.z, a2.w, a3.x, a3.y, a3.z, a3.w};

  // W2 is uniform: compiler promotes these to scalar loads.
  const f32x4* wrow = (const f32x4*)W2;
  f32x4 w0 = wrow[0], w1 = wrow[1], w2 = wrow[2], w3 = wrow[3];
  float w[16] = {w0.x, w0.y, w0.z, w0.w, w1.x, w1.y, w1.z, w1.w,
                 w2.x, w2.y, w2.z, w2.w, w3.x, w3.y, w3.z, w3.w};

  const float bias = b2[0];
  float acc[4];

#pragma unroll
  for (int jj = 0; jj < 4; ++jj) {
    const f32x4* brow = (const f32x4*)(Bb + (j0 + jj) * HIDDEN);
    f32x4 v0 = brow[0], v1 = brow[1], v2 = brow[2], v3 = brow[3];
    float bv[16] = {v0.x, v0.y, v0.z, v0.w, v1.x, v1.y, v1.z, v1.w,
                    v2.x, v2.y, v2.z, v2.w, v3.x, v3.y, v3.z, v3.w};
    float s = bias;
#pragma unroll
    for (int k = 0; k < 16; ++k) {
      s = fmaf(fmaxf(a[k] + bv[k], 0.0f), w[k], s);
    }
    acc[jj] = s;
  }

  f32x4 o = {acc[0], acc[1], acc[2], acc[3]};
  f32x4* dst = (f32x4*)(out + (size_t)i * N_NEURONS + j0);
  __builtin_nontemporal_store(o, dst);
}

extern "C" void kernel_launch(void* const* d_in, const int* in_sizes, int n_in,
                              void* d_out, int out_size, void* d_ws, size_t ws_size,
                              hipStream_t stream) {
  (void)in_sizes; (void)n_in; (void)out_size; (void)ws_size;
  const float* feat = (const float*)d_in[0];  // [4096, 3]
  const float* W1   = (const float*)d_in[1];  // [6, 16]
  const float* b1   = (const float*)d_in[2];  // [16]
  const float* W2   = (const float*)d_in[3];  // [16, 1]
  const float* b2   = (const float*)d_in[4];  // [1]
  float* out = (float*)d_out;                 // [4096, 4096]

  float* Abuf = (float*)d_ws;                 // 4096*16 f32 = 256 KB
  float* Bbuf = Abuf + N_NEURONS * HIDDEN;    // 4096*16 f32 = 256 KB

  nc_prologue_wmma<<<N_NEURONS / 16, 32, 0, stream>>>(feat, W1, b1, Abuf, Bbuf);

  dim3 grid(N_NEURONS / 32, N_NEURONS / 32);
  nc_pair_kernel<<<grid, 256, 0, stream>>>(Abuf, Bbuf, W2, b2, out);
}